// GeoDynamicLayer_17497696764666
// MI455X (gfx1250) — compile-verified
//
#include <hip/hip_runtime.h>
#include <hip/hip_bf16.h>

// Problem constants (from the reference)
#define DIN   2048
#define DOUT  4096
#define RANK  8
#define NB    8
#define NSEQ  2048

typedef __attribute__((ext_vector_type(2))) float v2f;
typedef __attribute__((ext_vector_type(4))) float v4f;
typedef __attribute__((ext_vector_type(8))) float v8f;
typedef __attribute__((ext_vector_type(4))) unsigned int u32x4;
typedef __attribute__((ext_vector_type(8))) int i32x8;
typedef __attribute__((ext_vector_type(4))) int i32x4;

static __device__ __forceinline__ v8f wmma_f32(v2f a, v2f b, v8f c) {
    // D = A(16x4) * B(4x16) + C(16x16), fp32 matrix path on CDNA5
    return __builtin_amdgcn_wmma_f32_16x16x4_f32(
        false, a, false, b, (short)0, c, false, false);
}

// ---------------------------------------------------------------------------
// TDM: issue one descriptor loading a (tile_dim1=128 rows) x (tile_dim0=16)
// f32 tile from a row-major tensor (row length 2048 elements) into LDS.
// Hardware inserts 4 pad DWORDs after every 16 data DWORDs -> LDS row stride
// of 20 floats (16B-aligned rows, conflict-free 16-row fragment reads).
// Descriptor packing per CDNA5 ISA sec. 8.3/8.4 (2D tile, groups 2/3 = 0).
// This toolchain exposes the 6-arg builtin (trailing group + cpol).
// ---------------------------------------------------------------------------
static __device__ __forceinline__ void tdm_load_tile_128x16(
    const float* gbase, unsigned lds_byte_off, unsigned rows_total) {
    unsigned long long ga = (unsigned long long)(__UINTPTR_TYPE__)gbase;
    u32x4 g0;
    g0[0] = 1u;                                   // count=1, user descriptor
    g0[1] = lds_byte_off;                         // lds_addr (bytes)
    g0[2] = (unsigned)ga;                         // global_addr[31:0]
    g0[3] = (unsigned)((ga >> 32) & 0x01FFFFFFu)  // global_addr[56:32]
            | (2u << 30);                         // type=2 ("image")
    i32x8 g1;
    g1[0] = (int)((2u << 16)     // data_size = 4 bytes
                | (1u << 20)     // pad_enable
                | (3u << 22)     // pad_interval: every 16 DWORDs
                | (3u << 25));   // pad_amount: 4 DWORDs
    g1[1] = (int)((2048u & 0xFFFFu) << 16);       // tensor_dim0 = 2048 (lo16)
    g1[2] = (int)((rows_total & 0xFFFFu) << 16);  // dim0 hi=0 | tensor_dim1 lo16
    g1[3] = (int)(16u << 16);                     // dim1 hi=0 | tile_dim0 = 16
    g1[4] = (int)128u;                            // tile_dim1 = 128, tile_dim2=0
    g1[5] = (int)2048u;                           // tensor_dim0_stride lo32
    g1[6] = 0;                                    // stride0 hi16 | stride1 lo16
    g1[7] = 0;                                    // stride1 hi32
    i32x4 z4 = {0, 0, 0, 0};
    i32x8 z8 = {0, 0, 0, 0, 0, 0, 0, 0};
    __builtin_amdgcn_tensor_load_to_lds(g0, g1, z4, z4, z8, 0);
}

// Per-lane async global->LDS b128 copy (ASYNCcnt path), inline asm so it is
// portable across toolchain builtin arities.
static __device__ __forceinline__ void async_copy_b128(unsigned lds_byte_off,
                                                       const float* gsrc) {
    asm volatile("global_load_async_to_lds_b128 %0, %1, off"
                 :: "v"(lds_byte_off), "v"(gsrc) : "memory");
}
static __device__ __forceinline__ void wait_async0() {
    asm volatile("s_wait_asynccnt 0" ::: "memory");
}

// ---------------------------------------------------------------------------
// Kernel 1a: partial sums over SEQ chunks (deterministic, no float atomics)
// ---------------------------------------------------------------------------
__global__ void k_meanpart(const float* __restrict__ x, float* __restrict__ zpart) {
    const int tid   = threadIdx.x;
    const int b     = blockIdx.x >> 3;
    const int i     = (blockIdx.x & 7) * 256 + tid;
    const int chunk = blockIdx.y;
    const int s0    = chunk * (NSEQ / 8);
    const float* px = x + (size_t)b * NSEQ * DIN + (size_t)s0 * DIN + i;
    float acc = 0.f;
    for (int s = 0; s < NSEQ / 8; ++s) acc += px[(size_t)s * DIN];
    zpart[(size_t)chunk * (NB * DIN) + b * DIN + i] = acc;
}

__global__ void k_meanred(const float* __restrict__ zpart, float* __restrict__ z) {
    const int tid = threadIdx.x;
    const int b   = blockIdx.x >> 3;
    const int i   = (blockIdx.x & 7) * 256 + tid;
    float acc = 0.f;
    #pragma unroll
    for (int c = 0; c < 8; ++c)
        acc += zpart[(size_t)c * (NB * DIN) + b * DIN + i];
    z[b * DIN + i] = acc * (1.0f / (float)NSEQ);
}

// ---------------------------------------------------------------------------
// Kernel 2: controller projections  P_w[b][j] = dot(z[b,:], W_w[j,:])
// ---------------------------------------------------------------------------
__global__ void k_proj(const float* __restrict__ z,
                       const float* __restrict__ Wa1, const float* __restrict__ Wa2,
                       const float* __restrict__ Wb1, const float* __restrict__ Wb2,
                       float* __restrict__ P) {
    const int tid = threadIdx.x;
    const int j   = blockIdx.x;
    const int w   = blockIdx.y;
    const float* W = (w == 0) ? Wa1 : (w == 1) ? Wa2 : (w == 2) ? Wb1 : Wb2;
    float acc[NB];
    #pragma unroll
    for (int b = 0; b < NB; ++b) acc[b] = 0.f;
    const float* wrow = W + (size_t)j * DIN;
    #pragma unroll
    for (int kk = 0; kk < DIN / 256; ++kk) {
        const int k = kk * 256 + tid;
        const float wv = wrow[k];
        #pragma unroll
        for (int b = 0; b < NB; ++b) acc[b] += wv * z[b * DIN + k];
    }
    __shared__ float red[256];
    float* Pw = P + (size_t)w * (NB * 16384);
    #pragma unroll
    for (int b = 0; b < NB; ++b) {
        red[tid] = acc[b];
        __syncthreads();
        for (int s = 128; s > 0; s >>= 1) {
            if (tid < s) red[tid] += red[tid + s];
            __syncthreads();
        }
        if (tid == 0) Pw[b * 16384 + j] = red[0];
        __syncthreads();
    }
}

// ---------------------------------------------------------------------------
// Kernel 3: low-rank outer products via WMMA, K = B*R = 64.
// ---------------------------------------------------------------------------
__global__ void k_lowrank(const float* __restrict__ Pa1, const float* __restrict__ Pa2,
                          const float* __restrict__ Pb1, const float* __restrict__ Pb2,
                          float* __restrict__ Araw, float* __restrict__ Bm) {
    const int lane = threadIdx.x & 31;
    const int wave = threadIdx.x >> 5;
    const int tile = blockIdx.x * 8 + wave;
    const int i0 = (tile >> 7) * 16;
    const int j0 = (tile & 127) * 16;
    const int lm = lane & 15;
    const int kh = (lane >> 4) * 2;

    v8f acc = {0.f,0.f,0.f,0.f,0.f,0.f,0.f,0.f};

    if (blockIdx.y == 0) {
        #pragma unroll
        for (int k0 = 0; k0 < 64; k0 += 4) {
            const int ka = k0 + kh, kb = ka + 1;
            v2f a, b;
            a.x = Pa1[(ka >> 3) * 16384 + (i0 + lm) * 8 + (ka & 7)];
            a.y = Pa1[(kb >> 3) * 16384 + (i0 + lm) * 8 + (kb & 7)];
            b.x = Pa2[(ka >> 3) * 16384 + (j0 + lm) * 8 + (ka & 7)];
            b.y = Pa2[(kb >> 3) * 16384 + (j0 + lm) * 8 + (kb & 7)];
            acc = wmma_f32(a, b, acc);
        }
        acc = acc * 0.125f;
        const int c = j0 + lm;
        const int r0 = i0 + ((lane >> 4) * 8);
        #pragma unroll
        for (int v = 0; v < 8; ++v) Araw[(size_t)(r0 + v) * DIN + c] = acc[v];
    } else {
        #pragma unroll
        for (int k0 = 0; k0 < 64; k0 += 4) {
            const int ka = k0 + kh, kb = ka + 1;
            v2f a, b;
            a.x = Pb1[(ka >> 3) * 16384 + (i0 + lm) * 8 + (ka & 7)];
            a.y = Pb1[(kb >> 3) * 16384 + (i0 + lm) * 8 + (kb & 7)];
            b.x = Pb2[(ka >> 3) * 16384 + (ka & 7) * DIN + (j0 + lm)];
            b.y = Pb2[(kb >> 3) * 16384 + (kb & 7) * DIN + (j0 + lm)];
            acc = wmma_f32(a, b, acc);
        }
        acc = acc * 0.125f;
        const int c = j0 + lm;
        const int r0 = i0 + ((lane >> 4) * 8);
        #pragma unroll
        for (int v = 0; v < 8; ++v) Bm[(size_t)(r0 + v) * DIN + c] = acc[v];
    }
}

// ---------------------------------------------------------------------------
// Kernel 4: skew-symmetrize  S = Araw - Araw^T
// ---------------------------------------------------------------------------
__global__ void k_skew(const float* __restrict__ Araw, float* __restrict__ S) {
    const int gid = blockIdx.x * 256 + threadIdx.x;
    const int i = gid >> 11;
    const int j = gid & 2047;
    S[gid] = Araw[gid] - Araw[(size_t)j * DIN + i];
}

// ---------------------------------------------------------------------------
// Kernel 5: W_dyn = U0 + U0 @ S + [0 ; B_mean]   (4096x2048 @ 2048x2048)
// LDS staging via async global->LDS (ASYNCcnt) instead of VGPR bounce.
// ---------------------------------------------------------------------------
__global__ void k_wdyn(const float* __restrict__ U0, const float* __restrict__ S,
                       const float* __restrict__ Bm, float* __restrict__ Wd) {
    __shared__ float Us[128 * 20];   // [m][k] stride 20
    __shared__ float Ss[16 * 132];   // [k][n] stride 132
    const int tid  = threadIdx.x;
    const int lane = tid & 31;
    const int wave = tid >> 5;
    const int wy = wave >> 2;
    const int wx = wave & 3;
    const int m0 = blockIdx.y * 128;
    const int n0 = blockIdx.x * 128;
    const int lm = lane & 15;
    const int kh = (lane >> 4) * 2;

    v8f acc[4][2];
    const v8f vz = {0.f,0.f,0.f,0.f,0.f,0.f,0.f,0.f};
    #pragma unroll
    for (int mt = 0; mt < 4; ++mt)
        #pragma unroll
        for (int nt = 0; nt < 2; ++nt) acc[mt][nt] = vz;

    for (int k0 = 0; k0 < DIN; k0 += 16) {
        #pragma unroll
        for (int it = 0; it < 2; ++it) {
            int g   = tid + it * 256;
            int row = g >> 2;
            int c4  = (g & 3) * 4;
            async_copy_b128((unsigned)(__UINTPTR_TYPE__)&Us[row * 20 + c4],
                            &U0[(size_t)(m0 + row) * DIN + k0 + c4]);
            int row2 = g >> 5;
            int c42  = (g & 31) * 4;
            async_copy_b128((unsigned)(__UINTPTR_TYPE__)&Ss[row2 * 132 + c42],
                            &S[(size_t)(k0 + row2) * DIN + n0 + c42]);
        }
        wait_async0();
        __syncthreads();
        #pragma unroll
        for (int kk = 0; kk < 16; kk += 4) {
            v2f a[4], b[2];
            #pragma unroll
            for (int mt = 0; mt < 4; ++mt) {
                const float* p = &Us[(wy * 64 + mt * 16 + lm) * 20 + kk + kh];
                a[mt].x = p[0]; a[mt].y = p[1];
            }
            #pragma unroll
            for (int nt = 0; nt < 2; ++nt) {
                const int n = wx * 32 + nt * 16 + lm;
                b[nt].x = Ss[(kk + kh) * 132 + n];
                b[nt].y = Ss[(kk + kh + 1) * 132 + n];
            }
            #pragma unroll
            for (int mt = 0; mt < 4; ++mt)
                #pragma unroll
                for (int nt = 0; nt < 2; ++nt)
                    acc[mt][nt] = wmma_f32(a[mt], b[nt], acc[mt][nt]);
        }
        __syncthreads();
    }

    const int rbase = m0 + wy * 64;
    const int cbase = n0 + wx * 32;
    #pragma unroll
    for (int mt = 0; mt < 4; ++mt) {
        #pragma unroll
        for (int nt = 0; nt < 2; ++nt) {
            const int r0 = rbase + mt * 16 + ((lane >> 4) * 8);
            const int c  = cbase + nt * 16 + lm;
            #pragma unroll
            for (int v = 0; v < 8; ++v) {
                const int r = r0 + v;
                float val = acc[mt][nt][v] + U0[(size_t)r * DIN + c];
                if (r >= DIN) val += Bm[(size_t)(r - DIN) * DIN + c];
                Wd[(size_t)r * DIN + c] = val;
            }
        }
    }
}

// ---------------------------------------------------------------------------
// Kernel 6: out[m, o] = sum_i X[m,i] * Wd[o,i]    (16384 x 4096 x K=2048)
// Double-buffered LDS, tiles DMA'd by the Tensor Data Mover (TENSORcnt);
// wave 0 drives the TDM, DMA of tile k+1 overlaps WMMA on tile k.
// ---------------------------------------------------------------------------
__global__ void k_final(const float* __restrict__ X, const float* __restrict__ Wd,
                        float* __restrict__ out) {
    __shared__ float Xs[2][128 * 20];   // [m][k] stride 20 (TDM pad)
    __shared__ float Ws[2][128 * 20];   // [o][k] stride 20 (TDM pad)
    const int tid  = threadIdx.x;
    const int lane = tid & 31;
    const int wave = tid >> 5;
    const int wy = wave >> 2;
    const int wx = wave & 3;
    const int m0 = blockIdx.y * 128;
    const int n0 = blockIdx.x * 128;
    const int lm = lane & 15;
    const int kh = (lane >> 4) * 2;

    v8f acc[4][2];
    const v8f vz = {0.f,0.f,0.f,0.f,0.f,0.f,0.f,0.f};
    #pragma unroll
    for (int mt = 0; mt < 4; ++mt)
        #pragma unroll
        for (int nt = 0; nt < 2; ++nt) acc[mt][nt] = vz;

    // Prologue: DMA first K-slab into buffer 0.
    if (wave == 0) {
        tdm_load_tile_128x16(&X[(size_t)m0 * DIN],
                             (unsigned)(__UINTPTR_TYPE__)&Xs[0][0], NB * NSEQ);
        tdm_load_tile_128x16(&Wd[(size_t)n0 * DIN],
                             (unsigned)(__UINTPTR_TYPE__)&Ws[0][0], DOUT);
    }

    for (int k0 = 0; k0 < DIN; k0 += 16) {
        const int buf = (k0 >> 4) & 1;
        if (wave == 0) {
            if (k0 + 16 < DIN) {
                // DMA next K-slab into the other buffer, then retire current:
                // tensor ops complete in order, so TENSORcnt<=2 means the two
                // loads for `buf` are done while the new pair is in flight.
                tdm_load_tile_128x16(&X[(size_t)m0 * DIN + (k0 + 16)],
                                     (unsigned)(__UINTPTR_TYPE__)&Xs[buf ^ 1][0],
                                     NB * NSEQ);
                tdm_load_tile_128x16(&Wd[(size_t)n0 * DIN + (k0 + 16)],
                                     (unsigned)(__UINTPTR_TYPE__)&Ws[buf ^ 1][0],
                                     DOUT);
                __builtin_amdgcn_s_wait_tensorcnt(2);
            } else {
                __builtin_amdgcn_s_wait_tensorcnt(0);
            }
        }
        __syncthreads();   // LDS buffer `buf` visible to all waves
        #pragma unroll
        for (int kk = 0; kk < 16; kk += 4) {
            v2f a[4], b[2];
            #pragma unroll
            for (int mt = 0; mt < 4; ++mt) {
                const float* p = &Xs[buf][(wy * 64 + mt * 16 + lm) * 20 + kk + kh];
                a[mt].x = p[0]; a[mt].y = p[1];
            }
            #pragma unroll
            for (int nt = 0; nt < 2; ++nt) {
                const float* p = &Ws[buf][(wx * 32 + nt * 16 + lm) * 20 + kk + kh];
                b[nt].x = p[0]; b[nt].y = p[1];
            }
            #pragma unroll
            for (int mt = 0; mt < 4; ++mt)
                #pragma unroll
                for (int nt = 0; nt < 2; ++nt)
                    acc[mt][nt] = wmma_f32(a[mt], b[nt], acc[mt][nt]);
        }
        __syncthreads();   // all reads of `buf` done before it is re-filled
    }

    const int rbase = m0 + wy * 64;
    const int cbase = n0 + wx * 32;
    #pragma unroll
    for (int mt = 0; mt < 4; ++mt) {
        #pragma unroll
        for (int nt = 0; nt < 2; ++nt) {
            const int r0 = rbase + mt * 16 + ((lane >> 4) * 8);
            const int c  = cbase + nt * 16 + lm;
            #pragma unroll
            for (int v = 0; v < 8; ++v)
                out[(size_t)(r0 + v) * DOUT + c] = acc[mt][nt][v];
        }
    }
}

// ---------------------------------------------------------------------------
// Workspace layout (floats): zpart@0, z@131072, P@147456, Araw@671744,
// S@4866048, Bm@9060352, Wd@13254656 ; total ~87 MB.
// ---------------------------------------------------------------------------
extern "C" void kernel_launch(void* const* d_in, const int* in_sizes, int n_in,
                              void* d_out, int out_size, void* d_ws, size_t ws_size,
                              hipStream_t stream) {
    const float* x   = (const float*)d_in[0];
    const float* U0  = (const float*)d_in[1];
    const float* Wa1 = (const float*)d_in[2];
    const float* Wa2 = (const float*)d_in[3];
    const float* Wb1 = (const float*)d_in[4];
    const float* Wb2 = (const float*)d_in[5];
    float* out = (float*)d_out;
    float* ws  = (float*)d_ws;

    float* zpart = ws;
    float* z     = ws + 131072;
    float* P     = ws + 147456;
    float* Araw  = ws + 671744;
    float* S     = ws + 4866048;
    float* Bm    = ws + 9060352;
    float* Wd    = ws + 13254656;

    k_meanpart<<<dim3(64, 8), 256, 0, stream>>>(x, zpart);
    k_meanred <<<64, 256, 0, stream>>>(zpart, z);
    k_proj    <<<dim3(16384, 4), 256, 0, stream>>>(z, Wa1, Wa2, Wb1, Wb2, P);
    k_lowrank <<<dim3(2048, 2), 256, 0, stream>>>(P, P + 131072, P + 262144,
                                                  P + 393216, Araw, Bm);
    k_skew    <<<16384, 256, 0, stream>>>(Araw, S);
    k_wdyn    <<<dim3(DIN / 128, DOUT / 128), 256, 0, stream>>>(U0, S, Bm, Wd);
    k_final   <<<dim3(DOUT / 128, (NB * NSEQ) / 128), 256, 0, stream>>>(x, Wd, out);
}